// NodeInfoScoreLayer_2388001817202
// MI455X (gfx1250) — compile-verified
//
#include <hip/hip_runtime.h>

typedef __attribute__((ext_vector_type(2))) float v2f;
typedef __attribute__((ext_vector_type(4))) float v4f;
typedef __attribute__((ext_vector_type(8))) float v8f;

#define FD 32  // feature dim

// ---------------------------------------------------------------------------
// 1) Zero the workspace (out_deg, in_deg, agg) with wide stores.
// ---------------------------------------------------------------------------
__global__ void gcn_zero_kernel(float* __restrict__ p, int n4) {
    int i = blockIdx.x * blockDim.x + threadIdx.x;
    if (i < n4) {
        v4f z = {0.0f, 0.0f, 0.0f, 0.0f};
        ((v4f*)p)[i] = z;
    }
}

// ---------------------------------------------------------------------------
// 2) Degree counts on the FULL graph (self-loops included, as in reference).
// ---------------------------------------------------------------------------
__global__ void gcn_degree_kernel(const int* __restrict__ src,
                                  const int* __restrict__ dst,
                                  unsigned* __restrict__ outd,
                                  unsigned* __restrict__ ind,
                                  int E) {
    int e = blockIdx.x * blockDim.x + threadIdx.x;
    if (e < E) {
        atomicAdd(&outd[src[e]], 1u);
        atomicAdd(&ind[dst[e]], 1u);
    }
}

// ---------------------------------------------------------------------------
// 3) Edge scatter: one wave32 per edge; lane = feature index.
//    msg = feat[src][d] * rsqrt(max(outdeg[src],1)) * e_feat[e]  (skip s==t)
//    Coalesced 128B feat-row gather + 32 contiguous f32 atomics to agg[dst].
// ---------------------------------------------------------------------------
__global__ void gcn_scatter_kernel(const float* __restrict__ feat,
                                   const float* __restrict__ e_feat,
                                   const int* __restrict__ src,
                                   const int* __restrict__ dst,
                                   const unsigned* __restrict__ outd,
                                   float* __restrict__ agg,
                                   int E) {
    int wave = blockIdx.x * (blockDim.x >> 5) + (threadIdx.x >> 5);
    int lane = threadIdx.x & 31;
    if (wave >= E) return;
    int s = src[wave];
    int t = dst[wave];
    if (s == t) return;  // remove_self_loop applies to aggregation only
    float w = e_feat[wave];
    unsigned dg = outd[s];
    float norm = rsqrtf((float)(dg ? dg : 1u));
    float v = feat[(size_t)s * FD + lane] * norm * w;
    atomicAdd(&agg[(size_t)t * FD + lane], v);
}

// ---------------------------------------------------------------------------
// 4) Finalize: one wave per 16-node tile.
//    val[r][c] = |feat[r][c] - agg[r][c] * rsqrt(max(indeg[r],1))|
//    rowsum via 8x V_WMMA_F32_16X16X4_F32 against an all-ones B matrix.
//    Lanes 0-15 own columns 0-15 of row (lane), lanes 16-31 own columns 16-31
//    of row (lane-16). With B == ones any K-permutation yields the row sum.
//    D-fragment: lane 0 holds rowsums of rows 0-7 (v0..v7), lane 16 rows 8-15.
// ---------------------------------------------------------------------------
__global__ void gcn_finalize_kernel(const float* __restrict__ feat,
                                    const float* __restrict__ agg,
                                    const unsigned* __restrict__ ind,
                                    float* __restrict__ out,
                                    int N) {
    int wave = blockIdx.x * (blockDim.x >> 5) + (threadIdx.x >> 5);
    int lane = threadIdx.x & 31;
    int ntiles = (N + 15) >> 4;
    if (wave >= ntiles) return;

    int r    = lane & 15;   // row within tile
    int half = lane >> 4;   // column half: 0 -> cols 0-15, 1 -> cols 16-31
    int node = wave * 16 + r;

    float vals[16];
    if (node < N) {
        unsigned dg = ind[node];
        float dn = rsqrtf((float)(dg ? dg : 1u));
        const v4f* fp = (const v4f*)(feat + (size_t)node * FD + half * 16);
        const v4f* ap = (const v4f*)(agg  + (size_t)node * FD + half * 16);
#pragma unroll
        for (int q = 0; q < 4; ++q) {
            v4f f = fp[q];
            v4f a = ap[q];
            vals[4 * q + 0] = __builtin_fabsf(f.x - a.x * dn);
            vals[4 * q + 1] = __builtin_fabsf(f.y - a.y * dn);
            vals[4 * q + 2] = __builtin_fabsf(f.z - a.z * dn);
            vals[4 * q + 3] = __builtin_fabsf(f.w - a.w * dn);
        }
    } else {
#pragma unroll
        for (int q = 0; q < 16; ++q) vals[q] = 0.0f;
    }

    v2f ones = {1.0f, 1.0f};
    v8f c = {};
#pragma unroll
    for (int kk = 0; kk < 8; ++kk) {
        v2f a;
        a.x = vals[2 * kk + 0];
        a.y = vals[2 * kk + 1];
        // 8 args: (neg_a, A, neg_b, B, c_mod, C, reuse_a, reuse_b)
        c = __builtin_amdgcn_wmma_f32_16x16x4_f32(
                false, a, false, ones, (short)0, c, false, false);
    }

    // D layout: lane n, VGPR m -> D[M][n]; lanes 0-15: M=m, lanes 16-31: M=m+8.
    // Column 0 lives in lane 0 (rows 0-7) and lane 16 (rows 8-15).
    if ((lane & 15) == 0) {
        int base = wave * 16 + half * 8;
#pragma unroll
        for (int m = 0; m < 8; ++m) {
            int nd = base + m;
            if (nd < N) out[nd] = c[m];
        }
    }
}

// ---------------------------------------------------------------------------
// Launch: inputs are (feat[N*32], e_feat[E], src[E], dst[E]); output [N] f32.
// Workspace: out_deg u32[N] | in_deg u32[N] | agg f32[N*32]  (~13.6 MB).
// ---------------------------------------------------------------------------
extern "C" void kernel_launch(void* const* d_in, const int* in_sizes, int n_in,
                              void* d_out, int out_size, void* d_ws, size_t ws_size,
                              hipStream_t stream) {
    const float* feat   = (const float*)d_in[0];
    const float* e_feat = (const float*)d_in[1];
    const int*   src    = (const int*)d_in[2];
    const int*   dst    = (const int*)d_in[3];
    float*       out    = (float*)d_out;

    int N = in_sizes[0] / FD;
    int E = in_sizes[2];

    unsigned* outd = (unsigned*)d_ws;
    unsigned* ind  = outd + N;
    float*    agg  = (float*)(ind + N);

    // 1) zero out_deg + in_deg + agg (all bit-zero)
    long long zero_words = (long long)2 * N + (long long)N * FD;
    int zero_v4 = (int)((zero_words + 3) / 4);
    gcn_zero_kernel<<<(zero_v4 + 255) / 256, 256, 0, stream>>>((float*)d_ws, zero_v4);

    // 2) degrees over the full edge list
    gcn_degree_kernel<<<(E + 255) / 256, 256, 0, stream>>>(src, dst, outd, ind, E);

    // 3) edge scatter: one wave per edge, 8 waves per 256-thread block
    gcn_scatter_kernel<<<(E + 7) / 8, 256, 0, stream>>>(feat, e_feat, src, dst,
                                                        outd, agg, E);

    // 4) finalize with WMMA row-sum: one wave per 16-node tile
    int ntiles = (N + 15) / 16;
    gcn_finalize_kernel<<<(ntiles + 7) / 8, 256, 0, stream>>>(feat, agg, ind, out, N);
}